// GCNBlock_78460462563622
// MI455X (gfx1250) — compile-verified
//
#include <hip/hip_runtime.h>
#include <hip/hip_bf16.h>

typedef float v2f __attribute__((ext_vector_type(2)));
typedef float v8f __attribute__((ext_vector_type(8)));

#define D_DIM   128
#define TILE_M  16
#define A_PAD   132   // 128 + 4 words: conflict-free ds_load_b64 per the bank math

// -------------------------------------------------------------------------
// h[n, o] = sum_d x[n,d] * W[o,d]   via V_WMMA_F32_16X16X4_F32 (fp32 WMMA)
//
// Per the CDNA5 ISA layouts (wave32):
//   A 16x4 f32 : lane l, vgpr v  holds A[l%16, 2*(l/16)+v]
//   B 4x16 f32 : lane l, vgpr v  holds B[2*(l/16)+v, l%16]
//   C/D 16x16  : lane l, vgpr v  holds D[v + 8*(l/16), l%16]
// Both A and B want k = 4*kt + 2*(l/16) + v  -> contiguous float2 loads.
// -------------------------------------------------------------------------
__global__ __launch_bounds__(256) void gemm_wmma_f32(
    const float* __restrict__ x, const float* __restrict__ W,
    float* __restrict__ h, int num_nodes)
{
    __shared__ __align__(16) float Atile[TILE_M * A_PAD];

    const int tid  = threadIdx.x;
    const int wave = tid >> 5;          // 0..7 -> owns cols [16*wave, 16*wave+16)
    const int lane = tid & 31;
    const int lm   = lane & 15;
    const int lg   = lane >> 4;
    const int n0   = wave * 16;

    // B = W^T slice for this wave's 16 output columns, kept in registers for
    // the whole kernel: 32 ktiles * float2 = 64 VGPRs per lane.
    v2f Breg[32];
    {
        const float* wrow = W + (size_t)(n0 + lm) * D_DIM + 2 * lg;
        #pragma unroll
        for (int kt = 0; kt < 32; ++kt)
            Breg[kt] = *(const v2f*)(wrow + 4 * kt);
    }

    const int num_tiles = (num_nodes + TILE_M - 1) / TILE_M;
    for (int t = blockIdx.x; t < num_tiles; t += gridDim.x) {
        const int row0 = t * TILE_M;

        // Stage the 16x128 A tile into padded LDS (coalesced global loads).
        #pragma unroll
        for (int j = 0; j < 8; ++j) {
            int idx = tid + 256 * j;          // 0..2047
            int r   = idx >> 7;
            int c   = idx & 127;
            int row = row0 + r;
            Atile[r * A_PAD + c] = (row < num_nodes)
                                     ? x[(size_t)row * D_DIM + c] : 0.0f;
        }
        __syncthreads();

        v8f acc = (v8f)0.0f;
        const float* Ap = Atile + lm * A_PAD + 2 * lg;
        #pragma unroll
        for (int kt = 0; kt < 32; ++kt) {
            v2f a = *(const v2f*)(Ap + 4 * kt);
            acc = __builtin_amdgcn_wmma_f32_16x16x4_f32(
                      /*neg_a=*/false, a,
                      /*neg_b=*/false, Breg[kt],
                      /*c_mod=*/(short)0, acc,
                      /*reuse_a=*/false, /*reuse_b=*/false);
        }
        __syncthreads();

        // Scatter C: lane l, vgpr v -> h[row0 + v + 8*lg, n0 + lm]
        float* hp = h + (size_t)(row0 + 8 * lg) * D_DIM + n0 + lm;
        #pragma unroll
        for (int v = 0; v < 8; ++v) {
            if (row0 + 8 * lg + v < num_nodes)
                hp[(size_t)v * D_DIM] = acc[v];
        }
    }
}

// -------------------------------------------------------------------------
// out[target[e], :] += h[source[e], :] * ew[e]   (32 lanes/edge, float4/lane)
// -------------------------------------------------------------------------
__global__ __launch_bounds__(256) void edge_scatter(
    const float* __restrict__ h, const float* __restrict__ ew,
    const int* __restrict__ src, const int* __restrict__ tgt,
    float* __restrict__ out, int num_edges)
{
    const long long total  = (long long)num_edges * 32;
    const long long stride = (long long)gridDim.x * blockDim.x;
    for (long long i = (long long)blockIdx.x * blockDim.x + threadIdx.x;
         i < total; i += stride) {
        int e = (int)(i >> 5);
        int p = ((int)i & 31) << 2;        // float4 slot
        float w = ew[e];
        const float4 hv = *(const float4*)(h + (size_t)src[e] * D_DIM + p);
        float* o = out + (size_t)tgt[e] * D_DIM + p;
        atomicAdd(o + 0, hv.x * w);
        atomicAdd(o + 1, hv.y * w);
        atomicAdd(o + 2, hv.z * w);
        atomicAdd(o + 3, hv.w * w);
    }
}

__global__ __launch_bounds__(256) void zero_f4(float4* __restrict__ out, int n4)
{
    const float4 z = make_float4(0.f, 0.f, 0.f, 0.f);
    int stride = gridDim.x * blockDim.x;
    for (int i = blockIdx.x * blockDim.x + threadIdx.x; i < n4; i += stride)
        out[i] = z;
}

// Fallback if the workspace can't hold h (no WMMA, but always correct path).
__global__ __launch_bounds__(128) void fused_fallback(
    const float* __restrict__ x, const float* __restrict__ W,
    const float* __restrict__ ew, const int* __restrict__ src,
    const int* __restrict__ tgt, float* __restrict__ out, int num_edges)
{
    for (int e = blockIdx.x; e < num_edges; e += gridDim.x) {
        int s = src[e], t = tgt[e];
        float w = ew[e];
        int o = threadIdx.x;   // output column
        const float* xr = x + (size_t)s * D_DIM;
        const float* wr = W + (size_t)o * D_DIM;
        float acc = 0.f;
        #pragma unroll 8
        for (int d = 0; d < D_DIM; ++d) acc += xr[d] * wr[d];
        atomicAdd(out + (size_t)t * D_DIM + o, acc * w);
    }
}

extern "C" void kernel_launch(void* const* d_in, const int* in_sizes, int n_in,
                              void* d_out, int out_size, void* d_ws, size_t ws_size,
                              hipStream_t stream)
{
    const float* x   = (const float*)d_in[0];   // (num_nodes, 128) f32
    const float* W   = (const float*)d_in[1];   // (128, 128) f32
    const float* ew  = (const float*)d_in[2];   // (num_edges,) f32
    const int*   src = (const int*)d_in[3];
    const int*   tgt = (const int*)d_in[4];
    float* out = (float*)d_out;

    const int num_nodes = in_sizes[0] / D_DIM;
    const int num_edges = in_sizes[2];

    // Output must be zeroed every call (harness poisons it).
    const int n4 = out_size / 4;
    zero_f4<<<1024, 256, 0, stream>>>((float4*)out, n4);

    const size_t need = (size_t)num_nodes * D_DIM * sizeof(float);
    if (ws_size >= need) {
        float* h = (float*)d_ws;
        gemm_wmma_f32<<<1024, 256, 0, stream>>>(x, W, h, num_nodes);
        edge_scatter<<<2048, 256, 0, stream>>>(h, ew, src, tgt, out, num_edges);
    } else {
        fused_fallback<<<4096, 128, 0, stream>>>(x, W, ew, src, tgt, out, num_edges);
    }
}